// STLossFun_17935783428750
// MI455X (gfx1250) — compile-verified
//
#include <hip/hip_runtime.h>

// ---------------------------------------------------------------------------
// STLoss (triplet) fused loss for MI455X / gfx1250 (wave32, WMMA).
//
//   loss_b = (e^{-0.5*t0} - e^{-||a-p||})^2 + (e^{-0.5*t1} - e^{-||a-n||})^2
//            + [Dap > Dan] * relu(v_an - v_ap)^2 ;  output = mean_b loss_b
//
// Bandwidth-bound (192 MiB of embeddings, ~0.7 flop/byte). Structure:
//   1. Block stages [32 rows x 64 K] tiles of a/p/n via fully-coalesced
//      global_load_b128 (each wave instruction = 2 rows x 256 B contiguous each,
//      full cache lines consumed exactly once).
//   2. Waves feed V_WMMA_F32_16X16X4_F32 from LDS: squared diffs land
//      directly in the documented f32 A-matrix layout (lane L -> row L&15,
//      K-slots (L>>4)*2..+1); B = all-ones so D[m][n] = rowsum(m).
//      LDS row stride 68 words (4*17, 17 coprime 16) => ds_load_b64 reads
//      are bank-conflict-free (half-waves hit disjoint bank classes mod 4).
//   3. global_prefetch of next round overlaps HBM latency with WMMA phase.
// ---------------------------------------------------------------------------

typedef float v2f __attribute__((ext_vector_type(2)));
typedef float v4f __attribute__((ext_vector_type(4)));
typedef float v8f __attribute__((ext_vector_type(8)));

#define B_ROWS           16384
#define DIM              1024
#define ROWS_PER_BLOCK   32            // 2 WMMA groups of 16 rows
#define KT               64            // K tile per staging round
#define NROUNDS          (DIM / KT)    // 16
#define ROWPAD           68            // 64 + 4 words: conflict-free A reads
#define NBLOCKS          (B_ROWS / ROWS_PER_BLOCK)   // 512

__global__ __launch_bounds__(256) void stloss_main_kernel(
    const float* __restrict__ ea,
    const float* __restrict__ ep,
    const float* __restrict__ en,
    const float* __restrict__ trip,      // [N,2]
    const int*   __restrict__ bidx,      // [B]
    float*       __restrict__ partial)   // [NBLOCKS]
{
    __shared__ float sA[ROWS_PER_BLOCK][ROWPAD];
    __shared__ float sP[ROWS_PER_BLOCK][ROWPAD];
    __shared__ float sN[ROWS_PER_BLOCK][ROWPAD];
    __shared__ float red_ap[2][4][16];
    __shared__ float red_an[2][4][16];

    const int tid  = threadIdx.x;
    const int lane = tid & 31;
    const int warp = tid >> 5;          // 0..7
    const int grp  = warp >> 2;         // 0..1 : 16-row group within block
    const int kq   = warp & 3;          // 0..3 : K quarter of each tile
    const int m    = lane & 15;         // A-matrix row M
    const int hi   = lane >> 4;         // 0: K-slots 0,1   1: K-slots 2,3

    const int rowbase = blockIdx.x * ROWS_PER_BLOCK;

    // Staging map: thread covers chunks {tid, tid+256}; 16 float4 per row.
    // Within one wave instruction: 2 rows x 256 B contiguous each.
    const int r0_st = tid >> 4;          // 0..15  (i=0)
    const int c_st  = (tid & 15) * 4;    // 0..60

    v8f acc_ap = {0.f, 0.f, 0.f, 0.f, 0.f, 0.f, 0.f, 0.f};
    v8f acc_an = {0.f, 0.f, 0.f, 0.f, 0.f, 0.f, 0.f, 0.f};
    const v2f ones = {1.0f, 1.0f};       // B matrix: 4x16 ones (layout-agnostic)

    const int arow = grp * 16 + m;       // LDS row this lane serves in WMMA

    for (int ro = 0; ro < NROUNDS; ++ro) {
        const int k0 = ro * KT;

        __syncthreads();                 // previous round's consume finished

        // ---- stage: coalesced global -> LDS (full cache lines) ----
#pragma unroll
        for (int i = 0; i < 2; ++i) {
            const int r = r0_st + 16 * i;
            const size_t g = (size_t)(rowbase + r) * DIM + k0 + c_st;
            v4f a4 = *(const v4f*)(ea + g);
            v4f p4 = *(const v4f*)(ep + g);
            v4f n4 = *(const v4f*)(en + g);
            *(v4f*)&sA[r][c_st] = a4;
            *(v4f*)&sP[r][c_st] = p4;
            *(v4f*)&sN[r][c_st] = n4;
        }
        __syncthreads();

        // ---- prefetch next round while WMMA phase runs ----
        if (ro + 1 < NROUNDS) {
#pragma unroll
            for (int i = 0; i < 2; ++i) {
                const size_t g = (size_t)(rowbase + r0_st + 16 * i) * DIM
                               + k0 + KT + c_st;
                __builtin_prefetch(ea + g, 0, 3);
                __builtin_prefetch(ep + g, 0, 3);
                __builtin_prefetch(en + g, 0, 3);
            }
        }

        // ---- consume: 16 K per wave per round, 4 x (K=4) WMMA steps ----
#pragma unroll
        for (int it = 0; it < 4; ++it) {
            const int k = kq * 16 + it * 4 + hi * 2;
            v2f av = *(const v2f*)&sA[arow][k];
            v2f pv = *(const v2f*)&sP[arow][k];
            v2f nv = *(const v2f*)&sN[arow][k];
            v2f dap = av - pv;  dap *= dap;     // squared diffs -> A matrix
            v2f dan = av - nv;  dan *= dan;
            acc_ap = __builtin_amdgcn_wmma_f32_16x16x4_f32(
                false, dap, false, ones, (short)0, acc_ap, false, false);
            acc_an = __builtin_amdgcn_wmma_f32_16x16x4_f32(
                false, dan, false, ones, (short)0, acc_an, false, false);
        }
    }

    // D layout: VGPR j = rowsum(j) in lanes 0-15, rowsum(j+8) in lanes 16-31
    // (columns replicated since B is all-ones). Lanes 0 and 16 export.
    if (m == 0) {
#pragma unroll
        for (int j = 0; j < 8; ++j) {
            red_ap[grp][kq][j + hi * 8] = acc_ap[j];
            red_an[grp][kq][j + hi * 8] = acc_an[j];
        }
    }
    __syncthreads();

    // Wave 0: one lane per row (32 rows). Combine K-quarters, scalar tail,
    // butterfly-reduce to one partial per block.
    if (warp == 0) {
        const int g = hi;                // lanes 0-15 -> group 0, 16-31 -> 1
        const int r = m;

        float s_ap = red_ap[g][0][r] + red_ap[g][1][r]
                   + red_ap[g][2][r] + red_ap[g][3][r];
        float s_an = red_an[g][0][r] + red_an[g][1][r]
                   + red_an[g][2][r] + red_an[g][3][r];

        float v_ap = expf(-sqrtf(s_ap));
        float v_an = expf(-sqrtf(s_an));

        const int grow = rowbase + g * 16 + r;
        const int idx  = bidx[grow];
        const float d0 = trip[2 * idx + 0] * 0.5f;   // EXTRA_COE
        const float d1 = trip[2 * idx + 1] * 0.5f;
        const float Dap = expf(-d0);
        const float Dan = expf(-d1);

        const float e0 = Dap - v_ap;
        const float e1 = Dan - v_an;
        float loss = e0 * e0 + e1 * e1;
        const float md = fmaxf(v_an - v_ap, 0.0f);   // relu
        if (Dap > Dan) loss += md * md;

#pragma unroll
        for (int off = 16; off > 0; off >>= 1)
            loss += __shfl_xor(loss, off, 32);

        if (lane == 0) partial[blockIdx.x] = loss;
    }
}

__global__ __launch_bounds__(512) void stloss_reduce_kernel(
    const float* __restrict__ partial, float* __restrict__ out,
    int n, float inv_b)
{
    __shared__ float sh[512];
    const int t = threadIdx.x;
    sh[t] = (t < n) ? partial[t] : 0.0f;
    __syncthreads();
#pragma unroll
    for (int s = 256; s > 0; s >>= 1) {
        if (t < s) sh[t] += sh[t + s];
        __syncthreads();
    }
    if (t == 0) out[0] = sh[0] * inv_b;
}

extern "C" void kernel_launch(void* const* d_in, const int* in_sizes, int n_in,
                              void* d_out, int out_size, void* d_ws, size_t ws_size,
                              hipStream_t stream) {
    const float* ea   = (const float*)d_in[0];   // embedding_a [16384,1024] f32
    const float* ep   = (const float*)d_in[1];   // embedding_p
    const float* en   = (const float*)d_in[2];   // embedding_n
    const float* trip = (const float*)d_in[3];   // triplets_dis [100000,2] f32
    const int*   bidx = (const int*)d_in[4];     // batch_index [16384]

    float* partial = (float*)d_ws;               // NBLOCKS floats of scratch
    float* out     = (float*)d_out;              // scalar f32

    stloss_main_kernel<<<NBLOCKS, 256, 0, stream>>>(ea, ep, en, trip, bidx, partial);
    stloss_reduce_kernel<<<1, 512, 0, stream>>>(partial, out, NBLOCKS,
                                                1.0f / (float)B_ROWS);
}